// AngularTripletLoss_19035295056420
// MI455X (gfx1250) — compile-verified
//
#include <hip/hip_runtime.h>

// ---------------------------------------------------------------------------
// Angular triplet loss on MI455X (gfx1250, wave32):
//   sim = E_all @ E_all^T  (fp32, only N anchor rows), arccos monotone =>
//   d_ap = acos(min positive sim), d_an = acos(max negative sim).
// Fused fp32 WMMA GEMM (V_WMMA_F32_16X16X4_F32) + masked min/max reduction,
// double-buffered LDS B tiles with async global->LDS staging when available.
// ---------------------------------------------------------------------------

typedef __attribute__((ext_vector_type(2))) float v2f;
typedef __attribute__((ext_vector_type(8))) float v8f;
// GCC-style vector to match the async builtin's parameter type exactly.
typedef int async_v4i __attribute__((vector_size(16)));

#define DIMK 128          // embedding dim
#define TILE 16           // WMMA tile M/N
#define WAVES_PER_WG 4    // 128 threads per workgroup
#define LDSB_STRIDE 132   // 128 + 4 pad: conflict-free banks, 16B aligned rows

#define GLOBAL_AS __attribute__((address_space(1)))
#define LDS_AS    __attribute__((address_space(3)))

#if defined(__has_builtin)
#if __has_builtin(__builtin_amdgcn_global_load_async_to_lds_b128) && \
    __has_builtin(__builtin_amdgcn_s_wait_asynccnt)
#define USE_ASYNC_LDS 1
#endif
#endif
#ifndef USE_ASYNC_LDS
#define USE_ASYNC_LDS 0
#endif

__device__ __forceinline__ unsigned encOrder(float f) {
    unsigned u = __float_as_uint(f);
    return (u & 0x80000000u) ? ~u : (u | 0x80000000u);
}
__device__ __forceinline__ float decOrder(unsigned u) {
    return (u & 0x80000000u) ? __uint_as_float(u & 0x7FFFFFFFu)
                             : __uint_as_float(~u);
}

__global__ void atl_init_kernel(unsigned* __restrict__ minpos,
                                unsigned* __restrict__ maxneg, int N) {
    int i = blockIdx.x * blockDim.x + threadIdx.x;
    if (i < N) {
        minpos[i] = encOrder(2.0f);   // sentinel: above any sim (<=1)
        maxneg[i] = encOrder(-2.0f);  // sentinel: below any sim (>=-1)
    }
}

// One wave owns a 16-anchor row tile (A fragments resident in 64 VGPRs);
// 4 waves/WG share double-buffered 16-col x 128 B tiles in LDS.
__global__ void __launch_bounds__(WAVES_PER_WG * 32)
atl_sim_kernel(const float* __restrict__ emb,      // [N,128] anchors
               const int*   __restrict__ labels,   // [N]
               const float* __restrict__ proto,    // [M,128]
               const int*   __restrict__ plabels,  // [M]
               unsigned* __restrict__ minpos,      // [N] order-encoded
               unsigned* __restrict__ maxneg,      // [N] order-encoded
               int N, int M, int colTilesPerWG) {
    const int nColTiles = (N + M) / TILE;

    const int lane = threadIdx.x & 31;
    const int wave = threadIdx.x >> 5;
    const int half = lane >> 4;   // 0: lanes 0-15, 1: lanes 16-31
    const int l16  = lane & 15;

    const int m0 = (blockIdx.x * WAVES_PER_WG + wave) * TILE; // anchor rows
    int ct       = blockIdx.y * colTilesPerWG;                // first col tile
    int ctEnd    = ct + colTilesPerWG;
    if (ctEnd > nColTiles) ctEnd = nColTiles;

    __shared__ float ldsB[2][TILE * LDSB_STRIDE];

    // Staging role of this thread: column sc, 64B chunk at skbase.
    const int sc     = threadIdx.x >> 3;        // 0..15
    const int skbase = (threadIdx.x & 7) * 16;  // 0,16,...,112

    // --- Preload A fragments (16x4 f32 layout: M=lane%16, K=k0+2*half) ----
    const float* aptr = emb + (size_t)(m0 + l16) * DIMK + 2 * half;
    v2f afrag[DIMK / 4];
#pragma unroll
    for (int kk = 0; kk < DIMK / 4; ++kk)
        afrag[kk] = *(const v2f*)(aptr + 4 * kk);

    // Row labels for the 8 rows this lane accumulates (C layout: M=v+8*half).
    int rl[8];
#pragma unroll
    for (int v = 0; v < 8; ++v) rl[v] = labels[m0 + v + 8 * half];

    float mp[8], mn[8];
#pragma unroll
    for (int v = 0; v < 8; ++v) { mp[v] = 2.0f; mn[v] = -2.0f; }

    // ---- Stage first B tile ------------------------------------------------
    {
        const int ng = ct * TILE + sc;
        const float* sp = ((ng < N) ? (emb + (size_t)ng * DIMK)
                                    : (proto + (size_t)(ng - N) * DIMK)) + skbase;
        float* dp = &ldsB[0][sc * LDSB_STRIDE + skbase];
#if USE_ASYNC_LDS
#pragma unroll
        for (int q = 0; q < 4; ++q)
            __builtin_amdgcn_global_load_async_to_lds_b128(
                (GLOBAL_AS async_v4i*)(sp + 4 * q),
                (LDS_AS async_v4i*)(dp + 4 * q), 0, 0);
        __builtin_amdgcn_s_wait_asynccnt(0);
#else
        float4 t0 = *(const float4*)(sp + 0);
        float4 t1 = *(const float4*)(sp + 4);
        float4 t2 = *(const float4*)(sp + 8);
        float4 t3 = *(const float4*)(sp + 12);
        *(float4*)(dp + 0)  = t0;
        *(float4*)(dp + 4)  = t1;
        *(float4*)(dp + 8)  = t2;
        *(float4*)(dp + 12) = t3;
#endif
    }
    __syncthreads();

    int p = 0;
    for (; ct < ctEnd; ++ct) {
        const bool hasNext = (ct + 1 < ctEnd);

        // ---- Issue next tile's staging (overlaps with compute below) ------
#if USE_ASYNC_LDS
        if (hasNext) {
            const int ng = (ct + 1) * TILE + sc;
            const float* sp = ((ng < N) ? (emb + (size_t)ng * DIMK)
                                        : (proto + (size_t)(ng - N) * DIMK)) + skbase;
            float* dp = &ldsB[p ^ 1][sc * LDSB_STRIDE + skbase];
#pragma unroll
            for (int q = 0; q < 4; ++q)
                __builtin_amdgcn_global_load_async_to_lds_b128(
                    (GLOBAL_AS async_v4i*)(sp + 4 * q),
                    (LDS_AS async_v4i*)(dp + 4 * q), 0, 0);
        }
#else
        float4 t0, t1, t2, t3;
        if (hasNext) {
            const int ng = (ct + 1) * TILE + sc;
            const float* sp = ((ng < N) ? (emb + (size_t)ng * DIMK)
                                        : (proto + (size_t)(ng - N) * DIMK)) + skbase;
            t0 = *(const float4*)(sp + 0);
            t1 = *(const float4*)(sp + 4);
            t2 = *(const float4*)(sp + 8);
            t3 = *(const float4*)(sp + 12);
        }
#endif

        // ---- Column label for this lane (issued early, used after WMMA) ---
        const int n0   = ct * TILE;
        const int ncol = n0 + l16;  // C layout: N = lane%16
        const int cl   = (ncol < N) ? labels[ncol] : plabels[ncol - N];

        // ---- 16x16 fp32 tile: 32 x V_WMMA_F32_16X16X4_F32, 2 accum chains -
        v8f acc0 = {}, acc1 = {};
        const float* bbase = &ldsB[p][l16 * LDSB_STRIDE + 2 * half];
#pragma unroll
        for (int kk = 0; kk < DIMK / 4; kk += 2) {
            v2f b0 = *(const v2f*)(bbase + 4 * kk);
            v2f b1 = *(const v2f*)(bbase + 4 * kk + 4);
            acc0 = __builtin_amdgcn_wmma_f32_16x16x4_f32(
                false, afrag[kk], false, b0, (short)0, acc0, false, false);
            acc1 = __builtin_amdgcn_wmma_f32_16x16x4_f32(
                false, afrag[kk + 1], false, b1, (short)0, acc1, false, false);
        }

        // ---- Masked running min(pos) / max(neg) ---------------------------
#pragma unroll
        for (int v = 0; v < 8; ++v) {
            const int   mrow = m0 + v + 8 * half;
            const float s    = acc0[v] + acc1[v];
            if (cl == rl[v]) {
                if (ncol != mrow) mp[v] = fminf(mp[v], s);  // exclude self
            } else {
                mn[v] = fmaxf(mn[v], s);
            }
        }

        // ---- Complete staging of the next tile, then barrier --------------
#if USE_ASYNC_LDS
        if (hasNext) __builtin_amdgcn_s_wait_asynccnt(0);
#else
        if (hasNext) {
            float* dp = &ldsB[p ^ 1][sc * LDSB_STRIDE + skbase];
            *(float4*)(dp + 0)  = t0;
            *(float4*)(dp + 4)  = t1;
            *(float4*)(dp + 8)  = t2;
            *(float4*)(dp + 12) = t3;
        }
#endif
        __syncthreads();
        p ^= 1;
    }

    // Reduce across the 16 lanes sharing the same rows (different columns).
#pragma unroll
    for (int v = 0; v < 8; ++v) {
#pragma unroll
        for (int off = 8; off >= 1; off >>= 1) {
            mp[v] = fminf(mp[v], __shfl_xor(mp[v], off, 16));
            mn[v] = fmaxf(mn[v], __shfl_xor(mn[v], off, 16));
        }
    }
    if (l16 == 0) {
#pragma unroll
        for (int v = 0; v < 8; ++v) {
            const int mrow = m0 + v + 8 * half;
            atomicMin(&minpos[mrow], encOrder(mp[v]));  // deterministic merge
            atomicMax(&maxneg[mrow], encOrder(mn[v]));
        }
    }
}

__global__ void atl_epilogue_kernel(const unsigned* __restrict__ minpos,
                                    const unsigned* __restrict__ maxneg,
                                    float* __restrict__ blockSum,
                                    float* __restrict__ blockCnt, int N) {
    __shared__ float ssum[256];
    __shared__ float scnt[256];
    const int i = blockIdx.x * blockDim.x + threadIdx.x;
    float per = 0.0f, cnt = 0.0f;
    if (i < N) {
        const float mp = decOrder(minpos[i]);
        const float mn = decOrder(maxneg[i]);
        const bool valid = (mp < 1.5f) && (mn > -1.5f);
        const float lo = -1.0f + 1e-7f, hi = 1.0f - 1e-7f;
        const float d_ap = acosf(fminf(fmaxf(mp, lo), hi));
        const float d_an = acosf(fminf(fmaxf(mn, lo), hi));
        const float pval = fmaxf(d_ap - d_an + 0.2f, 0.0f);
        if (valid) { per = pval; cnt = 1.0f; }
    }
    ssum[threadIdx.x] = per;
    scnt[threadIdx.x] = cnt;
    __syncthreads();
    for (int s = 128; s > 0; s >>= 1) {
        if (threadIdx.x < (unsigned)s) {
            ssum[threadIdx.x] += ssum[threadIdx.x + s];
            scnt[threadIdx.x] += scnt[threadIdx.x + s];
        }
        __syncthreads();
    }
    if (threadIdx.x == 0) {
        blockSum[blockIdx.x] = ssum[0];
        blockCnt[blockIdx.x] = scnt[0];
    }
}

__global__ void atl_finalize_kernel(const float* __restrict__ blockSum,
                                    const float* __restrict__ blockCnt,
                                    float* __restrict__ out, int nBlocks) {
    if (threadIdx.x == 0 && blockIdx.x == 0) {
        float s = 0.0f, c = 0.0f;
        for (int b = 0; b < nBlocks; ++b) { s += blockSum[b]; c += blockCnt[b]; }
        out[0] = (c > 0.0f) ? (s / c) : 0.0f;  // fixed-order: deterministic
    }
}

extern "C" void kernel_launch(void* const* d_in, const int* in_sizes, int n_in,
                              void* d_out, int out_size, void* d_ws,
                              size_t ws_size, hipStream_t stream) {
    const float* emb     = (const float*)d_in[0];  // [N,128] f32
    const int*   labels  = (const int*)d_in[1];    // [N] i32
    const float* proto   = (const float*)d_in[2];  // [M,128] f32
    const int*   plabels = (const int*)d_in[3];    // [M] i32
    float*       out     = (float*)d_out;

    const int N = in_sizes[1];  // 8192
    const int M = in_sizes[3];  // 2048

    // Workspace: minpos[N] u32 | maxneg[N] u32 | blockSum | blockCnt
    unsigned* minpos    = (unsigned*)d_ws;
    unsigned* maxneg    = minpos + N;
    const int epiBlocks = (N + 255) / 256;
    float* blockSum = (float*)(maxneg + N);
    float* blockCnt = blockSum + epiBlocks;

    atl_init_kernel<<<(N + 255) / 256, 256, 0, stream>>>(minpos, maxneg, N);

    const int nColTiles  = (N + M) / TILE;                      // 640
    const int nRowBlocks = N / (TILE * WAVES_PER_WG);           // 128
    const int nChunks    = 16;
    const int colTilesPerWG = (nColTiles + nChunks - 1) / nChunks;  // 40

    dim3 grid(nRowBlocks, nChunks);
    atl_sim_kernel<<<grid, WAVES_PER_WG * 32, 0, stream>>>(
        emb, labels, proto, plabels, minpos, maxneg, N, M, colTilesPerWG);

    atl_epilogue_kernel<<<epiBlocks, 256, 0, stream>>>(minpos, maxneg,
                                                       blockSum, blockCnt, N);
    atl_finalize_kernel<<<1, 1, 0, stream>>>(blockSum, blockCnt, out,
                                             epiBlocks);
}